// S4DKernel_26147760898121
// MI455X (gfx1250) — compile-verified
//
#include <hip/hip_runtime.h>
#include <math.h>

typedef float v2f __attribute__((ext_vector_type(2)));
typedef float v8f __attribute__((ext_vector_type(8)));

#define H_DIM   1024
#define NHALF   32
#define L_LEN   4096
#define T_BLK   64          // l = l0 + 64*l1 ; l0,l1 in [0,64)
#define USTR    68          // U row stride (floats): 68 % 64 == 4 -> banks {4L..4L+3}, conflict-free
#define VSTR    160         // Vp row stride (floats): 160 % 64 == 32 -> half-waves on disjoint banks

__global__ __launch_bounds__(256) void s4d_vandermonde_wmma(
    const float* __restrict__ log_dt,      // (H)
    const float* __restrict__ C_real,      // (H, 32, 2)
    const float* __restrict__ log_A_real,  // (H, 32)
    const float* __restrict__ A_imag,      // (H, 32)
    float* __restrict__ out)               // (H, 4096)
{
    // U[l1][k], k = 0..63 (re/im interleaved over n)            ~17.4 KB
    __shared__ float sU[T_BLK * USTR];
    // Vp[kp][2*l0 + {0,1}] = (Re w_n^l0, Im w_n^l0), kp == n    ~20.0 KB
    __shared__ float sVp[NHALF * VSTR];
    __shared__ float s_dAr[NHALF], s_dAi[NHALF];
    __shared__ float s_Cdr[NHALF], s_Cdi[NHALF];

    const int h = blockIdx.x;
    const int t = threadIdx.x;

    // ---- Stage 0: per-state scalars (dtA, Cd = C*(exp(dtA)-1)/A) ----------
    if (t < NHALF) {
        const int n = t;
        const float dt  = expf(log_dt[h]);
        const float Ar  = -expf(log_A_real[h * NHALF + n]);
        const float Ai  = A_imag[h * NHALF + n];
        const float dAr = Ar * dt;
        const float dAi = Ai * dt;
        const float er  = expf(dAr);
        float s0, c0;  sincosf(dAi, &s0, &c0);
        const float numr = er * c0 - 1.0f;
        const float numi = er * s0;
        const float inv = 1.0f / (Ar * Ar + Ai * Ai);
        const float qr  = (numr * Ar + numi * Ai) * inv;
        const float qi  = (numi * Ar - numr * Ai) * inv;
        const float Cr  = C_real[(h * NHALF + n) * 2 + 0];
        const float Ci  = C_real[(h * NHALF + n) * 2 + 1];
        s_Cdr[n] = Cr * qr - Ci * qi;
        s_Cdi[n] = Cr * qi + Ci * qr;
        s_dAr[n] = dAr;
        s_dAi[n] = dAi;
    }
    __syncthreads();

    // ---- Stage 1: build basis tables U (64x64) and Vp (32x(64x2)) in LDS --
    // Vp[n][2*l0+0] =  Re(w^l0),  Vp[n][2*l0+1] = Im(w^l0)
    // U[l1][2n]     =  Re(Cd*w^(64*l1)),  U[l1][2n+1] = -Im(Cd*w^(64*l1))
    for (int i = 0; i < 8; ++i) {
        const int e = t + 256 * i;           // 0..2047

        {   // V pair (contiguous b64 store)
            const int n  = e >> 6;
            const int l0 = e & 63;
            const float dAr = s_dAr[n], dAi = s_dAi[n];
            const float er = expf(dAr * (float)l0);
            float s, c;  sincosf(dAi * (float)l0, &s, &c);
            v2f v;  v.x = er * c;  v.y = er * s;
            *(v2f*)&sVp[n * VSTR + 2 * l0] = v;
        }
        {   // U pair (contiguous b64 store)
            const int n  = e & 31;
            const int l1 = e >> 5;
            const float dAr = s_dAr[n], dAi = s_dAi[n];
            const float p  = (float)(T_BLK * l1);
            const float er = expf(dAr * p);
            float s, c;  sincosf(dAi * p, &s, &c);
            const float wr = er * c, wi = er * s;
            const float Cdr = s_Cdr[n], Cdi = s_Cdi[n];
            v2f u;
            u.x =  (Cdr * wr - Cdi * wi);
            u.y = -(Cdr * wi + Cdi * wr);
            *(v2f*)&sU[l1 * USTR + 2 * n] = u;
        }
    }
    __syncthreads();

    // ---- Stage 2: K(64x64) = U x V via f32 WMMA 16x16x4 -------------------
    // Wave w: rows tile r = w>>1, column tiles c0 = 2*(w&1), c0+1 (shared A).
    const int wave = t >> 5;
    const int lane = t & 31;
    const int half = lane >> 4;      // 0: K={0,1}, 1: K={2,3} of each k-step
    const int mrow = lane & 15;      // A: row M ; B/D: col N
    const int r    = wave >> 1;
    const int c0   = (wave & 1) * 2;

    v8f acc0 = {}, acc1 = {};
    const float* Urow = &sU [(16 * r + mrow) * USTR + 2 * half];
    const float* Vb0  = &sVp[half * VSTR + 2 * (16 * c0 + mrow)];
    const float* Vb1  = Vb0 + 32;    // next column tile (+16 cols * 2 floats)

    #pragma unroll
    for (int kk = 0; kk < 16; ++kk) {
        v2f a  = *(const v2f*)(Urow + 4 * kk);          // K = 4kk+2h, +1
        v2f b0 = *(const v2f*)(Vb0 + 2 * kk * VSTR);    // kp = 2kk + half
        v2f b1 = *(const v2f*)(Vb1 + 2 * kk * VSTR);
        acc0 = __builtin_amdgcn_wmma_f32_16x16x4_f32(
                 false, a, false, b0, (short)0, acc0, false, false);
        acc1 = __builtin_amdgcn_wmma_f32_16x16x4_f32(
                 false, a, false, b1, (short)0, acc1, false, false);
    }

    // D layout: VGPR i -> M = i + 8*half, N = mrow
    float* outh = out + (size_t)h * L_LEN
                      + (size_t)(16 * r + 8 * half) * T_BLK
                      + 16 * c0 + mrow;
    #pragma unroll
    for (int i = 0; i < 8; ++i) {
        outh[i * T_BLK]      = 2.0f * acc0[i];
        outh[i * T_BLK + 16] = 2.0f * acc1[i];
    }
}

extern "C" void kernel_launch(void* const* d_in, const int* in_sizes, int n_in,
                              void* d_out, int out_size, void* d_ws, size_t ws_size,
                              hipStream_t stream) {
    (void)in_sizes; (void)n_in; (void)d_ws; (void)ws_size; (void)out_size;
    const float* log_dt     = (const float*)d_in[0];
    const float* C_real     = (const float*)d_in[1];
    const float* log_A_real = (const float*)d_in[2];
    const float* A_imag     = (const float*)d_in[3];
    // d_in[4] is the scalar L (=4096), baked in at compile time.
    float* out = (float*)d_out;

    s4d_vandermonde_wmma<<<H_DIM, 256, 0, stream>>>(
        log_dt, C_real, log_A_real, A_imag, out);
}